// DeepFitler_40218073759990
// MI455X (gfx1250) — compile-verified
//
#include <hip/hip_runtime.h>
#include <cstdint>

// DeepFilter: B=8, T=4096, F_TOTAL=481, NUM_FREQS=256, FRAME_SIZE=5, LOOKAHEAD=0
// spec : (8, 2, 4096, 481)  f32
// coefs: (8, 10, 4096, 256) f32   (k=0..4 real taps, k=5..9 imag taps)
// out  : (8, 2, 4096, 481)  f32
//
// Memory-bound (~590 MB @ 23.3 TB/s ~= 25 us). The spec tile (with 4-row time
// halo) is staged into LDS by the Tensor Data Mover (tensor_load_to_lds, one
// 2D-tile descriptor per complex component), tracked by TENSORcnt. Zero-reuse
// coefs/outputs stream with non-temporal hints.

#define NF      256
#define FT      481
#define TT      4096
#define NB      8
#define TILE_T  16
#define HALO    4               // FRAME_SIZE - 1
#define ROWS    (TILE_T + HALO) // 20
#define THREADS 256

typedef uint32_t u32x4 __attribute__((ext_vector_type(4)));
typedef uint32_t u32x8 __attribute__((ext_vector_type(8)));

// Issue one TDM 2D-tile load: tile_dim0 x tile_dim1 elements (4B each) cut out
// of a row-strided tensor at gaddr, packed contiguously into LDS at laddr.
__device__ __forceinline__ void tdm_load_2d_f32(uint32_t laddr, uint64_t gaddr,
                                                uint32_t tile_d0, uint32_t tile_d1,
                                                uint32_t tensor_d0, uint32_t row_stride) {
    u32x4 g0;
    g0[0] = 1u;                                    // count=1 (valid descriptor)
    g0[1] = laddr;                                 // lds_addr (bytes)
    g0[2] = (uint32_t)gaddr;                       // global_addr[31:0]
    g0[3] = (uint32_t)(gaddr >> 32) | (2u << 30);  // global_addr[56:32] | type=2

    u32x8 g1;
    g1[0] = 2u << 16;                              // data_size=2 (4 bytes); mask=0
    g1[1] = tensor_d0 << 16;                       // atomic_addr=0 | tensor_dim0 lo16
    g1[2] = (tensor_d0 >> 16) | (tile_d1 << 16);   // tensor_dim0 hi16 | tensor_dim1 lo16
    g1[3] = tile_d0 << 16;                         // tensor_dim1 hi16 | tile_dim0
    g1[4] = tile_d1;                               // tile_dim1 | tile_dim2=0 (unused)
    g1[5] = row_stride;                            // tensor_dim0_stride lo32
    g1[6] = row_stride << 16;                      // stride0 hi16=0 | stride1 lo16
    g1[7] = 0u;                                    // stride1 hi32 (unused for 2D)

    u32x4 gz = {0u, 0u, 0u, 0u};                   // groups 2/3: higher dims unused
    asm volatile("tensor_load_to_lds %0, %1, %2, %3"
                 :
                 : "s"(g0), "s"(g1), "s"(gz), "s"(gz)
                 : "memory");
}

__global__ void __launch_bounds__(THREADS)
deepfilter_kernel(const float* __restrict__ spec,
                  const float* __restrict__ coefs,
                  float* __restrict__ out) {
    // smem[comp][row][freq]; 2*20*256*4 = 40 KB
    __shared__ float smem[2 * ROWS * NF];

    const int b   = blockIdx.x;            // 0..7
    const int t0  = blockIdx.y * TILE_T;   // time tile start
    const int tid = threadIdx.x;           // 0..255 == frequency lane

    // ---- Stage spec[b][comp][t0-4 .. t0+TILE_T-1][0..255] into LDS via TDM --
    const bool first = (t0 == 0);
    const int  skip  = first ? HALO : 0;   // leading rows that are zero padding
    const int  nrows = ROWS - skip;        // rows fetched from memory
    const int  gt0   = t0 - HALO + skip;   // first valid global time row

    if (first) {
        // zero the 4 halo rows of both components (disjoint from TDM dest)
        for (int i = tid; i < HALO * NF; i += THREADS) {
            smem[i]             = 0.0f;    // comp 0, rows 0..3
            smem[ROWS * NF + i] = 0.0f;    // comp 1, rows 0..3
        }
    }

    if (tid < 32) {                        // wave 0 only: TDM ignores EXEC
#pragma unroll
        for (int comp = 0; comp < 2; ++comp) {
            const uint64_t gaddr = (uint64_t)(uintptr_t)
                (spec + (((size_t)(b * 2 + comp) * TT + gt0) * FT));
            const uint32_t laddr = (uint32_t)(uintptr_t)
                (&smem[(comp * ROWS + skip) * NF]);
            tdm_load_2d_f32(laddr, gaddr,
                            /*tile_d0=*/NF, /*tile_d1=*/(uint32_t)nrows,
                            /*tensor_d0=*/FT, /*row_stride=*/FT);
        }
        __builtin_amdgcn_s_wait_tensorcnt(0);  // both tiles landed in LDS
    }
    __syncthreads();

    // ---- Compute: 5-tap complex FIR per (t, f) ------------------------------
    const int    f      = tid;
    const size_t coef_b = (size_t)b * 10 * TT * NF;
    const size_t base_r = (size_t)(b * 2 + 0) * TT * FT;
    const size_t base_i = (size_t)(b * 2 + 1) * TT * FT;

    for (int tt = 0; tt < TILE_T; ++tt) {
        const int t = t0 + tt;
        float re = 0.0f, im = 0.0f;
#pragma unroll
        for (int k = 0; k < 5; ++k) {
            // coefs[b][k][t][f] / coefs[b][k+5][t][f] — zero reuse: stream NT
            const float cr = __builtin_nontemporal_load(
                coefs + coef_b + ((size_t)k * TT + t) * NF + f);
            const float ci = __builtin_nontemporal_load(
                coefs + coef_b + ((size_t)(k + 5) * TT + t) * NF + f);
            // smem row for global time (t-4+k) is (tt + k)
            const float pr = smem[(0 * ROWS + tt + k) * NF + f];
            const float pi = smem[(1 * ROWS + tt + k) * NF + f];
            re = fmaf(pr,  cr, re);
            re = fmaf(-pi, ci, re);
            im = fmaf(pi,  cr, im);
            im = fmaf(pr,  ci, im);
        }
        __builtin_nontemporal_store(re, out + base_r + (size_t)t * FT + f);
        __builtin_nontemporal_store(im, out + base_i + (size_t)t * FT + f);

        // Passthrough: f in [256, 481) — pure streaming copy
        if (tid < FT - NF) {
            const size_t off = (size_t)t * FT + NF + tid;
            __builtin_nontemporal_store(
                __builtin_nontemporal_load(spec + base_r + off), out + base_r + off);
            __builtin_nontemporal_store(
                __builtin_nontemporal_load(spec + base_i + off), out + base_i + off);
        }
    }
}

extern "C" void kernel_launch(void* const* d_in, const int* in_sizes, int n_in,
                              void* d_out, int out_size, void* d_ws, size_t ws_size,
                              hipStream_t stream) {
    const float* spec  = (const float*)d_in[0];
    const float* coefs = (const float*)d_in[1];
    float*       out   = (float*)d_out;
    (void)in_sizes; (void)n_in; (void)out_size; (void)d_ws; (void)ws_size;

    dim3 grid(NB, TT / TILE_T);   // (8, 256)
    dim3 block(THREADS);          // 256 threads = 8 wave32
    deepfilter_kernel<<<grid, block, 0, stream>>>(spec, coefs, out);
}